// GraphNN_52493090292366
// MI455X (gfx1250) — compile-verified
//
#include <hip/hip_runtime.h>
#include <hip/hip_bf16.h>
#include <stdint.h>

#define BOND_CUTOFF 3.6f
#define FEAT 7
#define TILE_J 1024
#define TILE_I 16
#define THREADS 256

typedef __attribute__((ext_vector_type(4))) float v4f;

// Stage one (x,y,z) triple (3 consecutive dwords in global) into LDS at the
// given LDS byte offset, using the CDNA5 async global->LDS copy engine.
// The instruction's immediate offset is added to BOTH the global source
// address and the LDS destination address (ISA 15.18.3), so offsets 0/4/8
// copy the 3 floats with matching strides on both sides.
__device__ __forceinline__ void async_copy_xyz(uint32_t lds_byte_off, const float* gsrc) {
    uint64_t g = (uint64_t)(uintptr_t)gsrc;
    asm volatile(
        "global_load_async_to_lds_b32 %0, %1, off\n\t"
        "global_load_async_to_lds_b32 %0, %1, off offset:4\n\t"
        "global_load_async_to_lds_b32 %0, %1, off offset:8"
        :
        : "v"(lds_byte_off), "v"(g)
        : "memory");
}

__device__ __forceinline__ void wait_async_all() {
    asm volatile("s_wait_asynccnt 0" ::: "memory");
}

__global__ __launch_bounds__(THREADS) void l1_bond_graph_kernel(
    const float* __restrict__ x, float* __restrict__ out, int N) {
    // Column-tile positions, padded to 4 floats/entry (16B) to avoid LDS bank conflicts.
    __shared__ float sPosJ[TILE_J * 4];   // 16 KB
    __shared__ float sPosI[TILE_I * 4];   // 256 B

    const int tid = threadIdx.x;
    const int j0  = blockIdx.x * TILE_J;
    const int i0  = blockIdx.y * TILE_I;

    // --- Stage column positions into LDS via the async copy engine ---
    #pragma unroll
    for (int k = 0; k < TILE_J / THREADS; ++k) {
        const int c = tid + k * THREADS;           // 0..1023
        const int j = j0 + c;
        if (j < N) {
            uint32_t loff = (uint32_t)(uintptr_t)(&sPosJ[c * 4]);
            async_copy_xyz(loff, x + (size_t)j * FEAT);
        }
    }
    // --- Stage row positions into LDS (first 16 lanes) ---
    if (tid < TILE_I) {
        const int i = i0 + tid;
        if (i < N) {
            uint32_t loff = (uint32_t)(uintptr_t)(&sPosI[tid * 4]);
            async_copy_xyz(loff, x + (size_t)i * FEAT);
        }
    }

    wait_async_all();
    __syncthreads();

    // --- Each thread owns 4 consecutive columns ---
    const int ct = tid * 4;          // local column base
    const int jt = j0 + ct;          // global column base

    float pjx[4], pjy[4], pjz[4];
    #pragma unroll
    for (int q = 0; q < 4; ++q) {
        pjx[q] = sPosJ[(ct + q) * 4 + 0];
        pjy[q] = sPosJ[(ct + q) * 4 + 1];
        pjz[q] = sPosJ[(ct + q) * 4 + 2];
    }

    const bool full_vec = (jt + 3) < N;

    #pragma unroll
    for (int ii = 0; ii < TILE_I; ++ii) {
        const int i = i0 + ii;
        if (i >= N) break;
        const float pix = sPosI[ii * 4 + 0];
        const float piy = sPosI[ii * 4 + 1];
        const float piz = sPosI[ii * 4 + 2];

        float r[4];
        #pragma unroll
        for (int q = 0; q < 4; ++q) {
            const float d = __builtin_fabsf(pix - pjx[q]) +
                            __builtin_fabsf(piy - pjy[q]) +
                            __builtin_fabsf(piz - pjz[q]);
            r[q] = (d <= BOND_CUTOFF) ? 1.0f : 0.0f;
        }

        float* row = out + (size_t)i * N;
        if (full_vec) {
            // Write-once 256MB stream > 192MB L2: use non-temporal hint so the
            // stores rinse toward HBM instead of cycling through L2 tags.
            // Coalesced 16B per lane -> 512B contiguous per wave32.
            v4f v = {r[0], r[1], r[2], r[3]};
            __builtin_nontemporal_store(v, (v4f*)(row + jt));
        } else {
            #pragma unroll
            for (int q = 0; q < 4; ++q)
                if (jt + q < N) __builtin_nontemporal_store(r[q], row + jt + q);
        }
    }
}

extern "C" void kernel_launch(void* const* d_in, const int* in_sizes, int n_in,
                              void* d_out, int out_size, void* d_ws, size_t ws_size,
                              hipStream_t stream) {
    const float* x = (const float*)d_in[0];
    float* out = (float*)d_out;
    const int N = in_sizes[0] / FEAT;   // 8192

    dim3 grid((N + TILE_J - 1) / TILE_J, (N + TILE_I - 1) / TILE_I);
    dim3 block(THREADS);
    l1_bond_graph_kernel<<<grid, block, 0, stream>>>(x, out, N);
}